// GeneAggregator_21019569946915
// MI455X (gfx1250) — compile-verified
//
#include <hip/hip_runtime.h>
#include <stdint.h>

// ---------------------------------------------------------------------------
// Segment-max gene aggregator for MI455X (gfx1250).
// B=16, V=16384, G=20000, D=64.  Memory-bound scatter-max:
//   pass 1: init d_out (as u32) to 0 (sentinel = "no variant hit this gene")
//   pass 2: stage embedding rows into LDS via async global->LDS b128 loads,
//           then per-row global_atomic_max_u32 on order-preserving encodings
//   pass 3: decode in place (0 -> 0.0f)
// ---------------------------------------------------------------------------

#define BLOCK          256        // 8 wave32 per block
#define WAVES_PER_BLK  (BLOCK / 32)
#define ROWS_PER_CHUNK 64         // 64 rows * 256B = 16KB LDS per block
#define ROW_BYTES      256        // D(=64) * sizeof(float)
#define EMB_D          64
#define BATCH_B        16

#if defined(__gfx1250__) && __has_builtin(__builtin_amdgcn_global_load_async_to_lds_b128)
#define USE_ASYNC_LDS 1
#else
#define USE_ASYNC_LDS 0
#endif

#if USE_ASYNC_LDS
// Parameter types per clang's diagnostic: pointer-to-v4i in global (AS1) and
// LDS (AS3) address spaces.
typedef int v4i_t __attribute__((vector_size(16)));
typedef v4i_t __attribute__((address_space(1))) *g_v4i_ptr;
typedef v4i_t __attribute__((address_space(3))) *l_v4i_ptr;
#endif

// Monotone (total-order preserving) float <-> u32 encoding.
// enc: positives -> bits ^ 0x80000000 ; negatives -> ~bits.
// Unsigned compare on enc == float compare. All finite floats encode > 0,
// so 0 is a safe "never touched" sentinel.
__device__ __forceinline__ unsigned encf(float f) {
    unsigned u = __float_as_uint(f);
    return u ^ ((unsigned)((int)u >> 31) | 0x80000000u);
}
__device__ __forceinline__ float decf(unsigned u) {
    unsigned m = (u & 0x80000000u) ? 0x80000000u : 0xFFFFFFFFu;
    return __uint_as_float(u ^ m);
}

// ---------------------------------------------------------------------------
__global__ void ga_init_kernel(uint4* __restrict__ out, long n4) {
    long i      = (long)blockIdx.x * blockDim.x + threadIdx.x;
    long stride = (long)gridDim.x * blockDim.x;
    uint4 z = make_uint4(0u, 0u, 0u, 0u);
    for (; i < n4; i += stride) out[i] = z;
}

// ---------------------------------------------------------------------------
__global__ __launch_bounds__(BLOCK) void ga_scatter_kernel(
    const float* __restrict__ emb,            // [R, 64] f32
    const int* __restrict__ gid,              // [R] int32 gene ids
    const unsigned char* __restrict__ mask,   // [R] bool (1 byte)
    const int* __restrict__ pG,               // scalar num_genes (device)
    unsigned* __restrict__ out,               // [B*G*64] encoded u32
    int R, int V)
{
    __shared__ __align__(16) unsigned char lds[ROWS_PER_CHUNK * ROW_BYTES];

    const int G            = *pG;
    const int chunkBaseRow = blockIdx.x * ROWS_PER_CHUNK;
    if (chunkBaseRow >= R) return;
    const int rowsHere  = min(ROWS_PER_CHUNK, R - chunkBaseRow);
    const int bytesHere = rowsHere * ROW_BYTES;

    // --- Stage chunk into LDS: 256 threads x 4 x 16B, coalesced ---
    const unsigned char* gbase =
        (const unsigned char*)emb + (size_t)chunkBaseRow * ROW_BYTES;
#pragma unroll
    for (int i = 0; i < (ROWS_PER_CHUNK * ROW_BYTES) / (BLOCK * 16); ++i) {
        int off = (i * BLOCK + threadIdx.x) * 16;
        if (off < bytesHere) {
#if USE_ASYNC_LDS
            __builtin_amdgcn_global_load_async_to_lds_b128(
                (g_v4i_ptr)(gbase + off),
                (l_v4i_ptr)(lds + off),
                0, 0);
#else
            *(uint4*)(lds + off) = *(const uint4*)(gbase + off);
#endif
        }
    }
#if USE_ASYNC_LDS
#if __has_builtin(__builtin_amdgcn_s_wait_asynccnt)
    __builtin_amdgcn_s_wait_asynccnt(0);
#else
    asm volatile("s_wait_asynccnt 0" ::: "memory");
#endif
#endif
    __syncthreads();

    // --- Scatter: one wave per row; lane l covers dims l and l+32 ---
    const int wave = threadIdx.x >> 5;
    const int lane = threadIdx.x & 31;
    for (int rr = wave; rr < rowsHere; rr += WAVES_PER_BLK) {
        const int r = chunkBaseRow + rr;
        if (!mask[r]) continue;
        const int g = gid[r];
        if (g < 0 || g >= G) continue;
        const int b = r / V;
        unsigned* dst = out + ((size_t)b * G + (size_t)g) * EMB_D;
        const float* lrow = (const float*)(lds + rr * ROW_BYTES);
        const unsigned e0 = encf(lrow[lane]);
        const unsigned e1 = encf(lrow[lane + 32]);
        atomicMax(dst + lane, e0);        // global_atomic_max_u32 (no return)
        atomicMax(dst + lane + 32, e1);
    }
}

// ---------------------------------------------------------------------------
__global__ void ga_finalize_kernel(uint4* __restrict__ out, long n4) {
    long i      = (long)blockIdx.x * blockDim.x + threadIdx.x;
    long stride = (long)gridDim.x * blockDim.x;
    for (; i < n4; i += stride) {
        uint4 u = out[i];
        float4 f;
        f.x = u.x ? decf(u.x) : 0.0f;
        f.y = u.y ? decf(u.y) : 0.0f;
        f.z = u.z ? decf(u.z) : 0.0f;
        f.w = u.w ? decf(u.w) : 0.0f;
        ((float4*)out)[i] = f;
    }
}

// ---------------------------------------------------------------------------
extern "C" void kernel_launch(void* const* d_in, const int* in_sizes, int n_in,
                              void* d_out, int out_size, void* d_ws, size_t ws_size,
                              hipStream_t stream) {
    const float*         emb  = (const float*)d_in[0];
    const int*           gid  = (const int*)d_in[1];
    const unsigned char* mask = (const unsigned char*)d_in[2];
    const int*           pG   = (const int*)d_in[3];
    unsigned*            out  = (unsigned*)d_out;

    const int  R  = in_sizes[1];      // B*V = 262144
    const int  V  = R / BATCH_B;      // 16384
    const long n4 = (long)out_size / 4;

    ga_init_kernel<<<2048, 256, 0, stream>>>((uint4*)d_out, n4);

    const int chunks = (R + ROWS_PER_CHUNK - 1) / ROWS_PER_CHUNK;  // 4096
    ga_scatter_kernel<<<chunks, BLOCK, 0, stream>>>(emb, gid, mask, pG, out, R, V);

    ga_finalize_kernel<<<2048, 256, 0, stream>>>((uint4*)d_out, n4);
}